// InnerAA_4063039062775
// MI455X (gfx1250) — compile-verified
//
#include <hip/hip_runtime.h>
#include <hip/hip_bf16.h>
#include <math.h>

#define N_TOK 3072
#define D_DIM 128
#define NUM_AA 20
#define NHEAD 8
#define HID   (2 * D_DIM)      // 256
#define NAH   (NHEAD * NUM_AA) // 160

typedef float v2f __attribute__((ext_vector_type(2)));
typedef float v8f __attribute__((ext_vector_type(8)));

__device__ __forceinline__ float gelu_tanh(float x) {
    // jax.nn.gelu (approximate=True)
    const float k0 = 0.7978845608028654f;
    const float k1 = 0.044715f;
    return 0.5f * x * (1.0f + tanhf(k0 * (x + k1 * x * x * x)));
}

// One wave32 accumulates a 16x16 f32 tile with V_WMMA_F32_16X16X4_F32.
// A: 16 rows of [.,K] row-major (lda), B: [K, n] row-major (ldb), 16-col slice.
// ISA 7.12.2 layouts: A lane l<16 holds (M=l, K=k+0/ k+1), lane>=16 holds K=k+2/k+3.
// B vgpr0 = rows K=k (lanes 0-15) / K=k+2 (lanes 16-31), vgpr1 = K=k+1 / K=k+3.
__device__ __forceinline__ v8f wmma_k_loop(const float* __restrict__ A, int lda,
                                           const float* __restrict__ B, int ldb,
                                           int K, v8f acc) {
    const int lane = threadIdx.x & 31;
    const int half = lane >> 4;
    const int l    = lane & 15;
    for (int k = 0; k < K; k += 4) {
        const int ka = k + 2 * half;
        v2f a, b;
        a.x = A[(size_t)l * lda + ka];
        a.y = A[(size_t)l * lda + ka + 1];
        b.x = B[(size_t)ka * ldb + l];
        b.y = B[(size_t)(ka + 1) * ldb + l];
        acc = __builtin_amdgcn_wmma_f32_16x16x4_f32(
            false, a, false, b, (short)0, acc, false, false);
    }
    return acc;
}

// C/D layout: lane l<16 element r -> (M=r, N=l); lane>=16 -> (M=r+8, N=l-16)
__device__ __forceinline__ void store_tile(float* __restrict__ C, int ldc,
                                           v8f acc, int epilogue) {
    const int lane = threadIdx.x & 31;
    const int half = lane >> 4;
    const int l    = lane & 15;
#pragma unroll
    for (int r = 0; r < 8; ++r) {
        float v = acc[r];
        if (epilogue == 1) v = gelu_tanh(v);
        C[(size_t)(r + 8 * half) * ldc + l] = v;
    }
}

// h = gelu(local @ W1[:128,:] + W1[128 + aa_gt[i], :])   (one-hot concat == row gather)
__global__ void __launch_bounds__(32)
k_hidden(const float* __restrict__ local, const int* __restrict__ aa_gt,
         const float* __restrict__ W1, float* __restrict__ h) {
    const int i0 = blockIdx.y * 16;
    const int c0 = blockIdx.x * 16;
    v8f acc = {};
    acc = wmma_k_loop(local + (size_t)i0 * D_DIM, D_DIM, W1 + c0, HID, D_DIM, acc);
    const int lane = threadIdx.x & 31;
    const int half = lane >> 4;
    const int l    = lane & 15;
#pragma unroll
    for (int r = 0; r < 8; ++r) {
        const int row = i0 + r + 8 * half;
        const int aa  = aa_gt[row];
        float v = acc[r] + W1[(size_t)(D_DIM + aa) * HID + c0 + l];
        h[(size_t)row * HID + c0 + l] = gelu_tanh(v);
    }
}

// generic C[M,n] = A[M,K] @ B[K,n], optional gelu epilogue
__global__ void __launch_bounds__(32)
k_gemm16(const float* __restrict__ A, const float* __restrict__ B,
         float* __restrict__ C, int K, int n, int epilogue) {
    const int i0 = blockIdx.y * 16;
    const int c0 = blockIdx.x * 16;
    v8f acc = {};
    acc = wmma_k_loop(A + (size_t)i0 * K, K, B + c0, n, K, acc);
    store_tile(C + (size_t)i0 * n + c0, n, acc, epilogue);
}

// tmat[j, a, x] = sum_b dgate[j, b, x] * inner[a, b, x]
__global__ void k_tmat(const float* __restrict__ dgate, const float* __restrict__ inner,
                       float* __restrict__ tmat) {
    const int idx = blockIdx.x * blockDim.x + threadIdx.x;
    if (idx >= N_TOK * NAH) return;
    const int j  = idx / NAH;
    const int ax = idx - j * NAH;
    const int a  = ax / NUM_AA;
    const int x  = ax - a * NUM_AA;
    float s = 0.f;
#pragma unroll
    for (int b = 0; b < NHEAD; ++b)
        s += dgate[(size_t)j * NAH + b * NUM_AA + x] * inner[(a * NHEAD + b) * NUM_AA + x];
    tmat[idx] = s;
}

// out[i,j,x] = log_softmax_x( sum_a dcode[i,a,x] * tmat[j,a,x] )
// One wave per 16x16 (i,j) tile; 20 v8f accumulators kept in registers.
__global__ void __launch_bounds__(32)
k_pair(const float* __restrict__ dcode, const float* __restrict__ tmat,
       float* __restrict__ out) {
    const int j0   = blockIdx.x * 16;
    const int i0   = blockIdx.y * 16;
    const int lane = threadIdx.x & 31;
    const int half = lane >> 4;
    const int l    = lane & 15;

    const float* __restrict__ Arow = dcode + (size_t)(i0 + l) * NAH; // A row i0+l
    const float* __restrict__ Bcol = tmat  + (size_t)(j0 + l) * NAH; // B column j0+l

    v8f acc[NUM_AA];
    const v8f zero = {};
#pragma unroll
    for (int x = 0; x < NUM_AA; ++x) acc[x] = zero;

#pragma unroll
    for (int x = 0; x < NUM_AA; ++x) {
#pragma unroll
        for (int k = 0; k < NHEAD; k += 4) {
            const int ka = k + 2 * half;
            v2f a, b;
            a.x = Arow[ka * NUM_AA + x];
            a.y = Arow[(ka + 1) * NUM_AA + x];
            b.x = Bcol[ka * NUM_AA + x];
            b.y = Bcol[(ka + 1) * NUM_AA + x];
            acc[x] = __builtin_amdgcn_wmma_f32_16x16x4_f32(
                false, a, false, b, (short)0, acc[x], false, false);
        }
    }

    // per-element log-softmax over the 20 x values, then coalesced float4 stores
    const int col = j0 + l;
#pragma unroll
    for (int r = 0; r < 8; ++r) {
        const int row = i0 + r + 8 * half;
        float m = -3.402823466e38f;
#pragma unroll
        for (int x = 0; x < NUM_AA; ++x) m = fmaxf(m, acc[x][r]);
        float s = 0.f;
#pragma unroll
        for (int x = 0; x < NUM_AA; ++x) s += __expf(acc[x][r] - m);
        const float lse = m + __logf(s);

        float4* dst = (float4*)(out + ((size_t)row * N_TOK + col) * NUM_AA);
#pragma unroll
        for (int q = 0; q < 5; ++q) {
            float4 o;
            o.x = acc[4 * q + 0][r] - lse;
            o.y = acc[4 * q + 1][r] - lse;
            o.z = acc[4 * q + 2][r] - lse;
            o.w = acc[4 * q + 3][r] - lse;
            dst[q] = o;
        }
    }
}

extern "C" void kernel_launch(void* const* d_in, const int* in_sizes, int n_in,
                              void* d_out, int out_size, void* d_ws, size_t ws_size,
                              hipStream_t stream) {
    const float* local = (const float*)d_in[0];
    const int*   aa_gt = (const int*)d_in[1];
    const float* W1    = (const float*)d_in[2];
    const float* W2    = (const float*)d_in[3];
    const float* Wc    = (const float*)d_in[4];
    const float* Wg    = (const float*)d_in[5];
    const float* inner = (const float*)d_in[6];
    float* out = (float*)d_out;

    float* ws    = (float*)d_ws;
    float* h     = ws;                               // N*256
    float* feat  = h     + (size_t)N_TOK * HID;      // N*128
    float* dcode = feat  + (size_t)N_TOK * D_DIM;    // N*160
    float* dgate = dcode + (size_t)N_TOK * NAH;      // N*160
    float* tmat  = dgate + (size_t)N_TOK * NAH;      // N*160  (total ~10.6 MB)

    const dim3 b32(32, 1, 1);
    k_hidden<<<dim3(HID / 16,  N_TOK / 16), b32, 0, stream>>>(local, aa_gt, W1, h);
    k_gemm16<<<dim3(D_DIM / 16, N_TOK / 16), b32, 0, stream>>>(h, W2, feat, HID, D_DIM, 0);
    k_gemm16<<<dim3(NAH / 16,   N_TOK / 16), b32, 0, stream>>>(feat, Wc, dcode, D_DIM, NAH, 0);
    k_gemm16<<<dim3(NAH / 16,   N_TOK / 16), b32, 0, stream>>>(feat, Wg, dgate, D_DIM, NAH, 1);

    const int t = N_TOK * NAH;
    k_tmat<<<(t + 255) / 256, 256, 0, stream>>>(dgate, inner, tmat);

    k_pair<<<dim3(N_TOK / 16, N_TOK / 16), b32, 0, stream>>>(dcode, tmat, out);
}